// BasicGCN_67989332295801
// MI455X (gfx1250) — compile-verified
//
#include <hip/hip_runtime.h>

typedef float v2f __attribute__((ext_vector_type(2)));
typedef float v8f __attribute__((ext_vector_type(8)));

// ---------------------------------------------------------------------------
// Degree / normalization: deg starts at 1.0 (self loop), count real edges,
// then dinv = rsqrt(deg) in place (deg >= 1 always, so no zero-guard needed).
// ---------------------------------------------------------------------------
__global__ void k_deg_init(float* __restrict__ deg, int n) {
    int i = blockIdx.x * blockDim.x + threadIdx.x;
    if (i < n) deg[i] = 1.0f;
}

__global__ void k_deg_count(const int* __restrict__ dst, float* __restrict__ deg, int e) {
    int i = blockIdx.x * blockDim.x + threadIdx.x;
    if (i < e)
        __hip_atomic_fetch_add(&deg[dst[i]], 1.0f, __ATOMIC_RELAXED, __HIP_MEMORY_SCOPE_AGENT);
}

__global__ void k_dinv(float* __restrict__ deg, int n) {
    int i = blockIdx.x * blockDim.x + threadIdx.x;
    if (i < n) deg[i] = rsqrtf(deg[i]);
}

// ---------------------------------------------------------------------------
// out[m][c] = bias[c]  (re-initialized every launch; harness does not re-poison)
// ---------------------------------------------------------------------------
__global__ void k_bias_init(float* __restrict__ out, const float* __restrict__ bias,
                            long total, int C) {
    long i = (long)blockIdx.x * blockDim.x + threadIdx.x;
    if (i < total) out[i] = bias[(int)(i % C)];
}

// ---------------------------------------------------------------------------
// C[M,N] = (RELU_A ? relu(A) : A)[M,K] @ B[K,N]  via V_WMMA_F32_16X16X4_F32.
//
// Block = 256 threads (8 waves). The whole block shares one 64-column B panel:
// it is staged once into LDS (64 KB for K=256), stored as (even-K, odd-K)
// float pairs so that each lane's B fragment for a k-step is ONE ds_load_b64
// at an immediate offset. Each wave then computes a 16-row strip x 64 cols
// (4 v8f accumulators), streaming A from global with a register double-buffer
// so the global load overlaps the 4 WMMAs of the previous k-step.
//
// Layouts per CDNA5 ISA 7.12.2 (wave32):
//   A 16x4 f32: lane l<16 -> row l, VGPR0=K0, VGPR1=K1; lanes 16..31 -> K2,K3.
//   B  4x16:    VGPR0 = {K0 lanes0-15, K2 lanes16-31}, VGPR1 = {K1, K3}, N=l.
//   C/D 16x16:  VGPR i = row (i + 8*half), col = l.
// Guards retire whole waves only => EXEC all-ones at every WMMA.
// ---------------------------------------------------------------------------
template <int K, bool RELU_A>
__global__ __launch_bounds__(256) void k_gemm_wmma(const float* __restrict__ A,
                                                   const float* __restrict__ B,
                                                   float* __restrict__ C,
                                                   int M, int N) {
    static_assert(K % 4 == 0, "K multiple of 4");
    __shared__ v2f ldsB[(K / 2) * 64];   // pair p holds {B[2p][c], B[2p+1][c]}

    const int tid   = threadIdx.x;
    const int lane  = tid & 31;
    const int wave  = tid >> 5;
    const int half  = lane >> 4;         // 0: lanes 0-15, 1: lanes 16-31
    const int l     = lane & 15;
    const int nBase = blockIdx.y * 64;

    // ---- cooperative B-panel stage: float4 rows -> paired LDS layout ----
    {
        const int c4 = (tid & 15) * 4;   // column group within the 64-col panel
        const int r0 = tid >> 4;         // starting row, 16 rows per pass
        for (int r = r0; r < K; r += 16) {
            float4 v = *(const float4*)&B[(size_t)r * N + nBase + c4];
            const int p = (r >> 1) * 64;
            const int j = r & 1;
            ldsB[p + c4 + 0][j] = v.x;
            ldsB[p + c4 + 1][j] = v.y;
            ldsB[p + c4 + 2][j] = v.z;
            ldsB[p + c4 + 3][j] = v.w;
        }
    }
    __syncthreads();

    const int mtile = blockIdx.x * 8 + wave;
    if (mtile >= (M >> 4)) return;       // uniform per wave; after the barrier
    const int m0 = mtile << 4;

    v8f acc[4];
#pragma unroll
    for (int t = 0; t < 4; ++t) acc[t] = (v8f){0.f, 0.f, 0.f, 0.f, 0.f, 0.f, 0.f, 0.f};

    const float* aptr = A + (size_t)(m0 + l) * K + 2 * half;  // lane's A stream
    const v2f*   bptr = &ldsB[half * 64 + l];                 // + (k/2)*64 + t*16

    v2f a = *(const v2f*)aptr;                                // k = 0 fragment
    if (RELU_A) { a.x = fmaxf(a.x, 0.0f); a.y = fmaxf(a.y, 0.0f); }

#pragma unroll 4
    for (int k = 0; k < K; k += 4) {
        v2f a_next = a;
        if (k + 4 < K) {                                      // prefetch next A
            a_next = *(const v2f*)(aptr + k + 4);
            if (RELU_A) { a_next.x = fmaxf(a_next.x, 0.0f); a_next.y = fmaxf(a_next.y, 0.0f); }
        }
        const v2f* bk = bptr + (k >> 1) * 64;
#pragma unroll
        for (int t = 0; t < 4; ++t) {
            v2f b = bk[t * 16];                               // ds_load_b64
            acc[t] = __builtin_amdgcn_wmma_f32_16x16x4_f32(
                /*neg_a=*/false, a, /*neg_b=*/false, b,
                /*c_mod=*/(short)0, acc[t], /*reuse_a=*/false, /*reuse_b=*/false);
        }
        a = a_next;
    }

#pragma unroll
    for (int t = 0; t < 4; ++t) {
        float* crow = C + (size_t)(m0 + 8 * half) * N + nBase + t * 16 + l;
#pragma unroll
        for (int i = 0; i < 8; ++i) crow[(size_t)i * N] = acc[t][i];
    }
}

// ---------------------------------------------------------------------------
// Scatter-add: one wave per (edge or self-loop) w in [0, E+Nnodes).
//   out[dst] += dinv[src]*dinv[dst] * H[src]   over C channels, VEC per lane.
// H rows live in L2 (100 MB < 192 MB), atomics resolve as f32 adds at L2.
// ---------------------------------------------------------------------------
template <int C, int VEC>
__global__ void k_scatter(const float* __restrict__ H, const int* __restrict__ src,
                          const int* __restrict__ dst, const float* __restrict__ dinv,
                          float* __restrict__ out, int E, int Nnodes) {
    const int  lane = threadIdx.x & 31;
    const long w    = (long)blockIdx.x * (blockDim.x >> 5) + (threadIdx.x >> 5);
    if (w >= (long)E + Nnodes) return;  // uniform per wave

    int s, d;
    if (w < E) { s = src[w]; d = dst[w]; }
    else       { s = d = (int)(w - E); }          // self loop

    const float  norm = dinv[s] * dinv[d];
    const float* hrow = H + (size_t)s * C;
    float*       orow = out + (size_t)d * C;

#pragma unroll
    for (int c = lane * VEC; c < C; c += 32 * VEC) {
        if (VEC == 4) {
            float4 h = *(const float4*)(hrow + c);
            __hip_atomic_fetch_add(orow + c + 0, norm * h.x, __ATOMIC_RELAXED, __HIP_MEMORY_SCOPE_AGENT);
            __hip_atomic_fetch_add(orow + c + 1, norm * h.y, __ATOMIC_RELAXED, __HIP_MEMORY_SCOPE_AGENT);
            __hip_atomic_fetch_add(orow + c + 2, norm * h.z, __ATOMIC_RELAXED, __HIP_MEMORY_SCOPE_AGENT);
            __hip_atomic_fetch_add(orow + c + 3, norm * h.w, __ATOMIC_RELAXED, __HIP_MEMORY_SCOPE_AGENT);
        } else {
            float2 h = *(const float2*)(hrow + c);
            __hip_atomic_fetch_add(orow + c + 0, norm * h.x, __ATOMIC_RELAXED, __HIP_MEMORY_SCOPE_AGENT);
            __hip_atomic_fetch_add(orow + c + 1, norm * h.y, __ATOMIC_RELAXED, __HIP_MEMORY_SCOPE_AGENT);
        }
    }
}

// ---------------------------------------------------------------------------
// In-place log_softmax over 64 channels; one wave32 per node, 2 chans/lane.
// ---------------------------------------------------------------------------
__global__ void k_log_softmax64(float* __restrict__ out, int n) {
    const int lane = threadIdx.x & 31;
    const int w    = blockIdx.x * (blockDim.x >> 5) + (threadIdx.x >> 5);
    if (w >= n) return;
    float* row = out + (size_t)w * 64;
    float v0 = row[lane];
    float v1 = row[lane + 32];
    float m = fmaxf(v0, v1);
#pragma unroll
    for (int off = 16; off > 0; off >>= 1) m = fmaxf(m, __shfl_xor(m, off));
    float s = expf(v0 - m) + expf(v1 - m);
#pragma unroll
    for (int off = 16; off > 0; off >>= 1) s += __shfl_xor(s, off);
    float lg = m + logf(s);
    row[lane]      = v0 - lg;
    row[lane + 32] = v1 - lg;
}

// ---------------------------------------------------------------------------
extern "C" void kernel_launch(void* const* d_in, const int* in_sizes, int n_in,
                              void* d_out, int out_size, void* d_ws, size_t ws_size,
                              hipStream_t stream) {
    (void)n_in; (void)out_size; (void)ws_size;

    const float* x    = (const float*)d_in[0];
    const int*   eidx = (const int*)d_in[1];
    const float* W1   = (const float*)d_in[2];
    const float* b1   = (const float*)d_in[3];
    const float* W2   = (const float*)d_in[4];
    const float* b2   = (const float*)d_in[5];
    float*       out  = (float*)d_out;

    const int DIN = 256, DHID = 256, DOUT = 64;
    const int n = in_sizes[0] / DIN;   // 100000
    const int E = in_sizes[1] / 2;     // 3200000
    const int* srcp = eidx;            // edge_index[0]
    const int* dstp = eidx + E;        // edge_index[1]

    // workspace layout: dinv | bufA (n*256) | bufB (n*256)   (~205.5 MB)
    char*  ws   = (char*)d_ws;
    float* dinv = (float*)ws;
    size_t off  = (((size_t)n * sizeof(float)) + 255) & ~(size_t)255;
    float* bufA = (float*)(ws + off);
    float* bufB = (float*)(ws + off + (size_t)n * DHID * sizeof(float));

    const int  mtiles = n / 16;        // n % 16 == 0 for this problem
    const long waves  = (long)E + n;   // edges + self loops

    // 1) degrees -> dinv
    k_deg_init<<<(n + 255) / 256, 256, 0, stream>>>(dinv, n);
    k_deg_count<<<(E + 255) / 256, 256, 0, stream>>>(dstp, dinv, E);
    k_dinv<<<(n + 255) / 256, 256, 0, stream>>>(dinv, n);

    // 2) H1 = x @ W1   (fp32 WMMA, LDS-staged B panel)
    {
        dim3 g((unsigned)((mtiles + 7) / 8), (unsigned)(DHID / 64));
        k_gemm_wmma<256, false><<<g, 256, 0, stream>>>(x, W1, bufA, n, DHID);
    }
    // 3) A1 = b1 ; scatter edges + self loops
    {
        long tot = (long)n * DHID;
        k_bias_init<<<(unsigned)((tot + 255) / 256), 256, 0, stream>>>(bufB, b1, tot, DHID);
        k_scatter<256, 4><<<(unsigned)((waves + 7) / 8), 256, 0, stream>>>(
            bufA, srcp, dstp, dinv, bufB, E, n);
    }
    // 4) H2 = relu(A1) @ W2  (ReLU fused into A stream), reuse bufA
    {
        dim3 g((unsigned)((mtiles + 7) / 8), (unsigned)(DOUT / 64));
        k_gemm_wmma<256, true><<<g, 256, 0, stream>>>(bufB, W2, bufA, n, DOUT);
    }
    // 5) out = b2 ; scatter layer 2
    {
        long tot = (long)n * DOUT;
        k_bias_init<<<(unsigned)((tot + 255) / 256), 256, 0, stream>>>(out, b2, tot, DOUT);
        k_scatter<64, 2><<<(unsigned)((waves + 7) / 8), 256, 0, stream>>>(
            bufA, srcp, dstp, dinv, out, E, n);
    }
    // 6) log_softmax in place
    k_log_softmax64<<<(n + 7) / 8, 256, 0, stream>>>(out, n);
}